// HybridBasisFunction_64355789963715
// MI455X (gfx1250) — compile-verified
//
#include <hip/hip_runtime.h>
#include <cstdint>

// ---------------------------------------------------------------------------
// HybridBasisFunction — elementwise, memory-bound (512 MiB @ 23.3 TB/s ≈ 23us).
// Reformulated as a 17-entry LDS lookup:  out = P[u] + Q[u]*w1,
//   u  = floor(clamp(x,0,1)*16)            (wavelet signs = bit parities of u)
//   w1 = clamp(x)*4 - (u>>2)               (spline fraction, exact in fp32)
// Data path: double-buffered async global->LDS (ASYNCcnt), NT b128 stores.
// ---------------------------------------------------------------------------

typedef float v4f __attribute__((ext_vector_type(4)));
typedef float v2f __attribute__((ext_vector_type(2)));
typedef int   g4i __attribute__((vector_size(16)));   // matches builtin param type

#define AS_GLOBAL __attribute__((address_space(1)))
#define AS_LOCAL  __attribute__((address_space(3)))

constexpr int TPB = 256;   // 8 wave32s per block
constexpr int IPT = 8;     // target float4-tiles per thread

__device__ __forceinline__ void wait_async_le1() {
  asm volatile("s_wait_asynccnt 0x1" ::: "memory");
}
__device__ __forceinline__ void wait_async_le0() {
  asm volatile("s_wait_asynccnt 0x0" ::: "memory");
}

#if __has_builtin(__builtin_amdgcn_global_load_async_to_lds_b128)
__device__ __forceinline__ void async_copy16(const v4f* g, v4f* l) {
  __builtin_amdgcn_global_load_async_to_lds_b128(
      (AS_GLOBAL g4i*)(uintptr_t)g,                 // global src (AS1 int4*)
      (AS_LOCAL  g4i*)(uint32_t)(uintptr_t)l,       // LDS dst: low 32 bits of flat ptr = LDS offset
      0, 0);                                        // imm offset, cpol
}
#else
__device__ __forceinline__ void async_copy16(const v4f* g, v4f* l) {
  unsigned loff = (unsigned)(uintptr_t)l;           // workgroup-relative LDS byte offset
  asm volatile("global_load_async_to_lds_b128 %0, %1, off"
               :: "v"(loff), "v"(g) : "memory");
}
#endif

__device__ __forceinline__ float hb_eval(float xv, const v2f* tab) {
  float xc  = __builtin_fminf(__builtin_fmaxf(xv, 0.0f), 1.0f);
  float t16 = xc * 16.0f;              // exact (power-of-2 scale)
  int   u   = (int)t16;                // trunc == floor for xc>=0; u in [0,16]
  v2f   pq  = tab[u];                  // ds_load_b64, conflict-free banks
  float fk  = (float)(u >> 2);         // spline knot index as float (exact)
  float w1  = __builtin_fmaf(xc, 4.0f, -fk);   // exact: xc*4 - k
  return __builtin_fmaf(pq.y, w1, pq.x);
}

__global__ __launch_bounds__(TPB) void HybridBasisFunction_kernel(
    const float* __restrict__ x, const float* __restrict__ wc,
    const float* __restrict__ sc, const float* __restrict__ alphap,
    float* __restrict__ out, long long n4, long long n)
{
  __shared__ v2f tab[20];              // 17 used
  __shared__ v4f stage[2][TPB];        // double-buffered async staging (8 KiB)

  const int tid = threadIdx.x;

  // Build the fused wavelet+spline table once per block (u = 0..16).
  if (tid < 17) {
    const int u = tid;
    const float sa = 1.0f / (1.0f + __expf(-alphap[0]));   // sigmoid(alpha)
    float W = wc[0];
    if (u < 16) {                       // u==16 <=> x>=1: all wavelet signs 0
      W += (u & 8) ? -wc[1] : wc[1];    // level 1: parity of floor(xc*2)
      W += (u & 4) ? -wc[2] : wc[2];    // level 2: parity of floor(xc*4)
      W += (u & 2) ? -wc[3] : wc[3];    // level 3: parity of floor(xc*8)
      W += (u & 1) ? -wc[4] : wc[4];    // level 4: parity of floor(xc*16)
    }
    const int k = u >> 2;               // 0..4
    const float A = sc[k];
    const float D = sc[k + 1] - sc[k];  // k+1 <= 5 < 8, no clamp needed
    v2f pq;
    pq.x = W + sa * A;                  // P[u]
    pq.y = sa * D;                      // Q[u]
    tab[u] = pq;
  }
  __syncthreads();

  const long long gstride = (long long)gridDim.x * TPB;
  const long long i0 = (long long)blockIdx.x * TPB + tid;
  const v4f* __restrict__ x4 = (const v4f*)x;
  v4f* __restrict__ o4 = (v4f*)out;

  // Steady-state: each thread streams float4 tiles through its private 16B
  // staging slot; per-wave ASYNCcnt orders its own async loads, so no
  // barriers are needed in the loop.
  int buf = 0;
  if (i0 < n4) async_copy16(x4 + i0, &stage[0][tid]);
  for (long long t = i0; t < n4; t += gstride) {
    const long long tn = t + gstride;
    if (tn < n4) {
      async_copy16(x4 + tn, &stage[buf ^ 1][tid]);
      wait_async_le1();                 // oldest (tile t) complete
    } else {
      wait_async_le0();                 // drain
    }
    v4f v = stage[buf][tid];            // ds_load_b128
    v4f r;
    r.x = hb_eval(v.x, tab);
    r.y = hb_eval(v.y, tab);
    r.z = hb_eval(v.z, tab);
    r.w = hb_eval(v.w, tab);
    __builtin_nontemporal_store(r, o4 + t);   // write-once: bypass L2 retention
    buf ^= 1;
  }

  // Scalar tail for n % 4 (zero for the reference shape, kept for generality).
  for (long long e = (n4 << 2) + i0; e < n; e += gstride) {
    out[e] = hb_eval(x[e], tab);
  }
}

extern "C" void kernel_launch(void* const* d_in, const int* in_sizes, int n_in,
                              void* d_out, int out_size, void* d_ws, size_t ws_size,
                              hipStream_t stream) {
  const float* x  = (const float*)d_in[0];   // (64, 4096, 256) fp32
  const float* wc = (const float*)d_in[1];   // 5 wavelet coeffs
  const float* sc = (const float*)d_in[2];   // 8 spline coeffs
  const float* al = (const float*)d_in[3];   // alpha scalar
  float* out = (float*)d_out;

  const long long n  = (long long)in_sizes[0];
  const long long n4 = n >> 2;

  long long blocks = (n4 + (long long)TPB * IPT - 1) / ((long long)TPB * IPT);
  if (blocks < 1) blocks = 1;
  if (blocks > 0x7FFFFFFFLL) blocks = 0x7FFFFFFFLL;

  HybridBasisFunction_kernel<<<dim3((unsigned)blocks), dim3(TPB), 0, stream>>>(
      x, wc, sc, al, out, n4, n);
}